// TestModule_3040836845632
// MI455X (gfx1250) — compile-verified
//
#include <hip/hip_runtime.h>
#include <hip/hip_bf16.h>

// ---------------- problem constants ----------------
#define N_NODES 50000
#define N_EDGES 800000
#define F_IN    64
#define HDIM    32
#define NCLS    10
#define CPAD    16               // classes padded to 16 for WMMA / aligned scatter
#define N_TILES (N_NODES / 16)   // 3125, exact

typedef __attribute__((ext_vector_type(2))) float v2f;
typedef __attribute__((ext_vector_type(8))) float v8f;

// workspace layout (floats)
#define WS_XW    0                          // x @ W1_rel            [N,32]
#define WS_AGG1  (WS_XW   + N_NODES*HDIM)   // scatter of xw         [N,32]
#define WS_H     (WS_AGG1 + N_NODES*HDIM)   // hidden activations    [N,32]
#define WS_HW    (WS_H    + N_NODES*HDIM)   // h @ W2_rel (padded)   [N,16]
#define WS_AGG2  (WS_HW   + N_NODES*CPAD)   // scatter of hw         [N,16]
#define WS_LOSS  (WS_AGG2 + N_NODES*CPAD)   // 1 float
#define WS_TOTAL (WS_LOSS + 1)

// ---------------- zero workspace ----------------
__global__ void gc_zero(float* __restrict__ ws, int n) {
    int i = blockIdx.x * blockDim.x + threadIdx.x;
    if (i < n) ws[i] = 0.0f;
}

// ---------------- GEMM: out[N,32] = in[N,64] @ W[64,32]  (f32 WMMA) -----------
__global__ __launch_bounds__(32) void gc_gemm64x32(
        const float* __restrict__ in, const float* __restrict__ W,
        float* __restrict__ out) {
    int tile = blockIdx.x;
    int lane = threadIdx.x;
    int hi   = lane >> 4;
    int col0 = lane & 15;
    int row  = tile * 16 + col0;
    const float* arow = in + (size_t)row * F_IN;

    v8f acc0 = {}; v8f acc1 = {};
    #pragma unroll
    for (int s = 0; s < 16; ++s) {
        int k = 4 * s + 2 * hi;
        v2f a  = { arow[k], arow[k + 1] };
        v2f b0 = { W[k * HDIM + col0],      W[(k + 1) * HDIM + col0] };
        v2f b1v= { W[k * HDIM + col0 + 16], W[(k + 1) * HDIM + col0 + 16] };
        acc0 = __builtin_amdgcn_wmma_f32_16x16x4_f32(false, a, false, b0, (short)0, acc0, false, false);
        acc1 = __builtin_amdgcn_wmma_f32_16x16x4_f32(false, a, false, b1v, (short)0, acc1, false, false);
    }
    #pragma unroll
    for (int v = 0; v < 8; ++v) {
        int r = tile * 16 + v + 8 * hi;
        out[(size_t)r * HDIM + col0]      = acc0[v];
        out[(size_t)r * HDIM + col0 + 16] = acc1[v];
    }
}

// ---------------- scatter 1: agg1[dst] += xw[src]  (32 f32/edge) --------------
// thread t: edge = t>>1, 16-float half = (t&1)*16
__global__ void gc_scatter32(const float* __restrict__ xw,
                             const int* __restrict__ ei,
                             float* __restrict__ agg1) {
    int t = blockIdx.x * blockDim.x + threadIdx.x;
    int e = t >> 1;
    if (e >= N_EDGES) return;
    int c = (t & 1) * 16;
    int s = ei[e];
    int d = ei[N_EDGES + e];
    const float4* xs = (const float4*)(xw + (size_t)s * HDIM + c);
    float* dp = agg1 + (size_t)d * HDIM + c;
    #pragma unroll
    for (int i = 0; i < 4; ++i) {
        float4 v = xs[i];
        unsafeAtomicAdd(dp + 4 * i + 0, v.x);
        unsafeAtomicAdd(dp + 4 * i + 1, v.y);
        unsafeAtomicAdd(dp + 4 * i + 2, v.z);
        unsafeAtomicAdd(dp + 4 * i + 3, v.w);
    }
}

// ---------------- layer 1 tail: h = relu(x@W1root + agg1 + b1) ----------------
__global__ __launch_bounds__(32) void gc_layer1_tail(
        const float* __restrict__ x, const float* __restrict__ W1root,
        const float* __restrict__ agg1, const float* __restrict__ b1,
        float* __restrict__ h) {
    int tile = blockIdx.x;
    int lane = threadIdx.x;
    int hi   = lane >> 4;
    int col0 = lane & 15;
    int row  = tile * 16 + col0;
    const float* xrow = x + (size_t)row * F_IN;

    v8f acc0 = {}; v8f acc1 = {};
    #pragma unroll
    for (int s = 0; s < 16; ++s) {
        int k = 4 * s + 2 * hi;
        v2f a  = { xrow[k], xrow[k + 1] };
        v2f b0 = { W1root[k * HDIM + col0],      W1root[(k + 1) * HDIM + col0] };
        v2f b1v= { W1root[k * HDIM + col0 + 16], W1root[(k + 1) * HDIM + col0 + 16] };
        acc0 = __builtin_amdgcn_wmma_f32_16x16x4_f32(false, a, false, b0, (short)0, acc0, false, false);
        acc1 = __builtin_amdgcn_wmma_f32_16x16x4_f32(false, a, false, b1v, (short)0, acc1, false, false);
    }
    float bias0 = b1[col0];
    float bias1 = b1[col0 + 16];
    #pragma unroll
    for (int v = 0; v < 8; ++v) {
        int r = tile * 16 + v + 8 * hi;
        float o0 = acc0[v] + agg1[(size_t)r * HDIM + col0]      + bias0;
        float o1 = acc1[v] + agg1[(size_t)r * HDIM + col0 + 16] + bias1;
        o0 = o0 > 0.0f ? o0 : 0.0f;
        o1 = o1 > 0.0f ? o1 : 0.0f;
        h[(size_t)r * HDIM + col0]      = o0;
        h[(size_t)r * HDIM + col0 + 16] = o1;
    }
}

// ---------------- GEMM: hw[N,16] = h[N,32] @ W2rel(pad to 16 cols) ------------
__global__ __launch_bounds__(32) void gc_gemm32x16(
        const float* __restrict__ h, const float* __restrict__ W2rel,
        float* __restrict__ hw) {
    __shared__ float Wlds[HDIM * CPAD];   // [32][16], cols >= 10 zero
    int lane = threadIdx.x;
    for (int i = lane; i < HDIM * CPAD; i += 32) {
        int r = i >> 4, c = i & 15;
        Wlds[i] = (c < NCLS) ? W2rel[r * NCLS + c] : 0.0f;
    }
    __syncthreads();

    int tile = blockIdx.x;
    int hi   = lane >> 4;
    int col0 = lane & 15;
    int row  = tile * 16 + col0;
    const float* hrow = h + (size_t)row * HDIM;

    v8f acc = {};
    #pragma unroll
    for (int s = 0; s < 8; ++s) {
        int k = 4 * s + 2 * hi;
        v2f a = { hrow[k], hrow[k + 1] };
        v2f b = { Wlds[k * CPAD + col0], Wlds[(k + 1) * CPAD + col0] };
        acc = __builtin_amdgcn_wmma_f32_16x16x4_f32(false, a, false, b, (short)0, acc, false, false);
    }
    #pragma unroll
    for (int v = 0; v < 8; ++v) {
        int r = tile * 16 + v + 8 * hi;
        hw[(size_t)r * CPAD + col0] = acc[v];   // pad cols are exactly 0
    }
}

// ---------------- scatter 2: agg2[dst] += hw[src]  (16 f32/edge) --------------
__global__ void gc_scatter16(const float* __restrict__ hw,
                             const int* __restrict__ ei,
                             float* __restrict__ agg2) {
    int e = blockIdx.x * blockDim.x + threadIdx.x;
    if (e >= N_EDGES) return;
    int s = ei[e];
    int d = ei[N_EDGES + e];
    const float4* hs = (const float4*)(hw + (size_t)s * CPAD);
    float* dp = agg2 + (size_t)d * CPAD;
    #pragma unroll
    for (int i = 0; i < 4; ++i) {
        float4 v = hs[i];
        unsafeAtomicAdd(dp + 4 * i + 0, v.x);
        unsafeAtomicAdd(dp + 4 * i + 1, v.y);
        unsafeAtomicAdd(dp + 4 * i + 2, v.z);
        unsafeAtomicAdd(dp + 4 * i + 3, v.w);
    }
}

// ---------------- layer 2 tail + cross-entropy --------------------------------
// logits = h@W2root + agg2 + b2 ; cols 10..15 -> -inf ; logsumexp - true_logit
__global__ __launch_bounds__(32) void gc_layer2_loss(
        const float* __restrict__ h, const float* __restrict__ W2root,
        const float* __restrict__ agg2, const float* __restrict__ b2,
        const int* __restrict__ y, float* __restrict__ lossAcc) {
    __shared__ float Wlds[HDIM * CPAD];   // [32][16] padded W2root
    __shared__ float tlds[16 * 16];       // logits tile, row-major

    int lane = threadIdx.x;
    for (int i = lane; i < HDIM * CPAD; i += 32) {
        int r = i >> 4, c = i & 15;
        Wlds[i] = (c < NCLS) ? W2root[r * NCLS + c] : 0.0f;
    }
    __syncthreads();

    int tile = blockIdx.x;
    int hi   = lane >> 4;
    int col0 = lane & 15;
    int row  = tile * 16 + col0;
    const float* hrow = h + (size_t)row * HDIM;

    v8f acc = {};
    #pragma unroll
    for (int s = 0; s < 8; ++s) {
        int k = 4 * s + 2 * hi;
        v2f a = { hrow[k], hrow[k + 1] };
        v2f b = { Wlds[k * CPAD + col0], Wlds[(k + 1) * CPAD + col0] };
        acc = __builtin_amdgcn_wmma_f32_16x16x4_f32(false, a, false, b, (short)0, acc, false, false);
    }

    float bias = (col0 < NCLS) ? b2[col0] : 0.0f;
    #pragma unroll
    for (int v = 0; v < 8; ++v) {
        int r = v + 8 * hi;
        int gr = tile * 16 + r;
        float val = acc[v] + agg2[(size_t)gr * CPAD + col0] + bias;
        if (col0 >= NCLS) val = -1e30f;
        tlds[r * 16 + col0] = val;
    }
    __syncthreads();

    float contrib = 0.0f;
    if (lane < 16) {
        int node = tile * 16 + lane;
        const float* lg = tlds + lane * 16;
        float m = lg[0];
        #pragma unroll
        for (int c = 1; c < NCLS; ++c) m = fmaxf(m, lg[c]);
        float sum = 0.0f;
        #pragma unroll
        for (int c = 0; c < NCLS; ++c) sum += expf(lg[c] - m);
        float logZ = m + logf(sum);
        contrib = logZ - lg[y[node]];
    }
    #pragma unroll
    for (int off = 16; off > 0; off >>= 1)
        contrib += __shfl_down(contrib, off, 32);
    if (lane == 0) unsafeAtomicAdd(lossAcc, contrib);
}

// ---------------- finalize ----------------
__global__ void gc_finalize(const float* __restrict__ lossAcc, float* __restrict__ out) {
    if (threadIdx.x == 0 && blockIdx.x == 0)
        out[0] = lossAcc[0] * (1.0f / (float)N_NODES);
}

extern "C" void kernel_launch(void* const* d_in, const int* in_sizes, int n_in,
                              void* d_out, int out_size, void* d_ws, size_t ws_size,
                              hipStream_t stream) {
    const float* x      = (const float*)d_in[0];
    const int*   ei     = (const int*)  d_in[1];
    const int*   y      = (const int*)  d_in[2];
    const float* W1rel  = (const float*)d_in[3];
    const float* b1     = (const float*)d_in[4];
    const float* W1root = (const float*)d_in[5];
    const float* W2rel  = (const float*)d_in[6];
    const float* b2     = (const float*)d_in[7];
    const float* W2root = (const float*)d_in[8];
    float* out = (float*)d_out;
    float* ws  = (float*)d_ws;

    float* xw   = ws + WS_XW;
    float* agg1 = ws + WS_AGG1;
    float* hbuf = ws + WS_H;
    float* hw   = ws + WS_HW;
    float* agg2 = ws + WS_AGG2;
    float* loss = ws + WS_LOSS;

    // 1) zero workspace (agg buffers + loss; rest overwritten anyway)
    gc_zero<<<(WS_TOTAL + 255) / 256, 256, 0, stream>>>(ws, WS_TOTAL);
    // 2) xw = x @ W1_rel   (push nodes through lin_rel BEFORE the scatter)
    gc_gemm64x32<<<N_TILES, 32, 0, stream>>>(x, W1rel, xw);
    // 3) agg1[dst] += xw[src]   (32 f32/edge instead of 64)
    {
        int nt = N_EDGES * 2;
        gc_scatter32<<<(nt + 255) / 256, 256, 0, stream>>>(xw, ei, agg1);
    }
    // 4) h = relu(x @ W1_root + agg1 + b1)
    gc_layer1_tail<<<N_TILES, 32, 0, stream>>>(x, W1root, agg1, b1, hbuf);
    // 5) hw = h @ W2_rel (padded to 16 cols)
    gc_gemm32x16<<<N_TILES, 32, 0, stream>>>(hbuf, W2rel, hw);
    // 6) agg2[dst] += hw[src]   (16 f32/edge instead of 32)
    gc_scatter16<<<(N_EDGES + 255) / 256, 256, 0, stream>>>(hw, ei, agg2);
    // 7) logits = h @ W2_root + agg2 + b2, fused cross-entropy
    gc_layer2_loss<<<N_TILES, 32, 0, stream>>>(hbuf, W2root, agg2, b2, y, loss);
    // 8) mean
    gc_finalize<<<1, 32, 0, stream>>>(loss, out);
}